// HANNodeAttention_13013750906994
// MI455X (gfx1250) — compile-verified
//
#include <hip/hip_runtime.h>
#include <hip/hip_bf16.h>
#include <cmath>

// HAN node attention, MI455X (gfx1250, wave32).
// Memory-bound problem (~770MB HBM traffic, ~33us at 23.3TB/s); compute is
// negligible, so keep exact f32 math via V_WMMA_F32_16X16X4_F32.

#define N_NODES 8192
#define IN_DIM  512
#define H_DIM   128

typedef __attribute__((ext_vector_type(2))) float v2f;
typedef __attribute__((ext_vector_type(4))) float f4;
typedef __attribute__((ext_vector_type(8))) float v8f;

__device__ __forceinline__ float leaky02(float x) { return x > 0.f ? x : 0.2f * x; }

// ---------------------------------------------------------------------------
// Kernel 1: Wh = h @ W_w^T + b   (8192x512x128 GEMM, f32 WMMA)
//           s  = Wh @ a          (fused via LDS float atomics)
// Block = 256 threads (8 waves); block owns 16 rows, wave w owns cols 16w.
// ---------------------------------------------------------------------------
__global__ __launch_bounds__(256) void wh_kernel(const float* __restrict__ h,
                                                 const float* __restrict__ Ww,
                                                 const float* __restrict__ Wb,
                                                 const float* __restrict__ avec,
                                                 float* __restrict__ Wh,
                                                 float* __restrict__ s) {
  __shared__ float lds_h[16][IN_DIM + 4];   // pad 4 -> conflict-free A reads
  __shared__ float lds_s[16];
  const int tid = threadIdx.x;
  const int r0 = blockIdx.x * 16;

  if (tid < 16) lds_s[tid] = 0.f;
  for (int idx = tid; idx < 16 * IN_DIM; idx += 256) {
    const int r = idx >> 9, c = idx & (IN_DIM - 1);
    lds_h[r][c] = h[(size_t)(r0 + r) * IN_DIM + c];
  }
  __syncthreads();

  const int wave = tid >> 5, lane = tid & 31;
  const int n0 = wave * 16;
  const int mrow = lane & 15;             // A rows / D cols index
  const int kb = (lane < 16) ? 0 : 2;     // A-layout: half-waves hold K, K+2
  const int n = n0 + mrow;
  const float* wrow = Ww + (size_t)n * IN_DIM;  // B[k][n] = W_w[n][k]

  v8f acc = {};
  for (int k0 = 0; k0 < IN_DIM; k0 += 4) {
    v2f af = { lds_h[mrow][k0 + kb], lds_h[mrow][k0 + kb + 1] };
    v2f bf = { wrow[k0 + kb],        wrow[k0 + kb + 1] };
    acc = __builtin_amdgcn_wmma_f32_16x16x4_f32(false, af, false, bf,
                                                (short)0, acc, false, false);
  }

  const int mb = (lane < 16) ? 0 : 8;     // C/D layout: VGPR p = rows p, p+8
  const float bias = Wb[n];
  const float an = avec[n];
#pragma unroll
  for (int p = 0; p < 8; ++p) {
    const float val = acc[p] + bias;
    Wh[(size_t)(r0 + mb + p) * H_DIM + n] = val;
    atomicAdd(&lds_s[mb + p], val * an);  // ds_add_f32
  }
  __syncthreads();
  if (tid < 16) s[r0 + tid] = lds_s[tid];
}

// ---------------------------------------------------------------------------
// Kernel 2: masked leaky-relu row softmax -> alpha.  One block per row.
// Thread t owns 32 contiguous columns; mask kept as a register bitfield so
// adj is streamed exactly once (nontemporal), alpha written once (NT).
// ---------------------------------------------------------------------------
__global__ __launch_bounds__(256) void alpha_kernel(const float* __restrict__ adj,
                                                    const float* __restrict__ s,
                                                    float* __restrict__ alpha) {
  __shared__ float red[256];
  const int tid = threadIdx.x;
  const int i = blockIdx.x;
  const float si = s[i];
  const float* arow = adj + (size_t)i * N_NODES;
  float* orow = alpha + (size_t)i * N_NODES;
  const int j0 = tid * 32;

  // pass 1: neighbor mask + masked row max
  unsigned mbits = 0;
  float lm = -3.402823466e38f;
#pragma unroll
  for (int c = 0; c < 8; ++c) {
    const f4 av = __builtin_nontemporal_load((const f4*)(arow + j0 + 4 * c));
    const f4 sv = *(const f4*)(s + j0 + 4 * c);
#pragma unroll
    for (int q = 0; q < 4; ++q) {
      if (av[q] != 0.f) {
        mbits |= (1u << (4 * c + q));
        lm = fmaxf(lm, leaky02(si + sv[q]));
      }
    }
  }
  red[tid] = lm;
  __syncthreads();
  for (int off = 128; off; off >>= 1) {
    if (tid < off) red[tid] = fmaxf(red[tid], red[tid + off]);
    __syncthreads();
  }
  const float m = red[0];
  __syncthreads();

  // pass 2: exp-sum (recompute from s in L1/L2; cheaper than re-streaming)
  float ls = 0.f;
#pragma unroll
  for (int c = 0; c < 8; ++c) {
    const f4 sv = *(const f4*)(s + j0 + 4 * c);
#pragma unroll
    for (int q = 0; q < 4; ++q)
      if (mbits & (1u << (4 * c + q)))
        ls += __expf(leaky02(si + sv[q]) - m);
  }
  red[tid] = ls;
  __syncthreads();
  for (int off = 128; off; off >>= 1) {
    if (tid < off) red[tid] += red[tid + off];
    __syncthreads();
  }
  const float denom = red[0];
  const float inv = (denom > 0.f) ? 1.f / denom : 0.f;  // isolated rows -> 0

  // pass 3: normalize + nontemporal store
#pragma unroll
  for (int c = 0; c < 8; ++c) {
    const f4 sv = *(const f4*)(s + j0 + 4 * c);
    f4 ov;
#pragma unroll
    for (int q = 0; q < 4; ++q)
      ov[q] = (mbits & (1u << (4 * c + q)))
                  ? __expf(leaky02(si + sv[q]) - m) * inv
                  : 0.f;
    __builtin_nontemporal_store(ov, (f4*)(orow + j0 + 4 * c));
  }
}

// ---------------------------------------------------------------------------
// Kernel 3: z = sigmoid(alpha @ Wh)   (8192x8192x128 GEMM, f32 WMMA)
// Block = 256 threads; block owns 64 rows x 128 cols. Wave (rg,cg) owns a
// 16x64 strip = 4 accumulator tiles. alpha + Wh chunks staged in LDS (51KB).
// ---------------------------------------------------------------------------
#define KC 64
__global__ __launch_bounds__(256) void z_kernel(const float* __restrict__ alpha,
                                                const float* __restrict__ Wh,
                                                float* __restrict__ z) {
  __shared__ float lds_a[64][KC + 4];      // pad 4: conflict-free A-frag reads
  __shared__ float lds_w[KC][H_DIM + 4];
  const int tid = threadIdx.x;
  const int r64 = blockIdx.x * 64;
  const int wave = tid >> 5, lane = tid & 31;
  const int rg = wave & 3;                 // row group (16 rows each)
  const int cg = wave >> 2;                // col group (64 cols each)
  const int mrow = lane & 15;
  const int kb = (lane < 16) ? 0 : 2;

  v8f acc[4] = {};
  for (int k0 = 0; k0 < N_NODES; k0 += KC) {
    // stage alpha[r64:r64+64][k0:k0+KC)  (single-use -> nontemporal)
    for (int idx = tid; idx < 64 * (KC / 4); idx += 256) {
      const int r = idx / (KC / 4), c4 = idx % (KC / 4);
      const f4 v = __builtin_nontemporal_load(
          (const f4*)(alpha + (size_t)(r64 + r) * N_NODES + k0 + 4 * c4));
      *(f4*)&lds_a[r][4 * c4] = v;
    }
    // stage Wh[k0:k0+KC)[0:128)  (L2 resident)
    for (int idx = tid; idx < KC * (H_DIM / 4); idx += 256) {
      const int kk = idx / (H_DIM / 4), c4 = idx % (H_DIM / 4);
      const f4 v = *(const f4*)(Wh + (size_t)(k0 + kk) * H_DIM + 4 * c4);
      *(f4*)&lds_w[kk][4 * c4] = v;
    }
    __syncthreads();

    if (lane == 0)  // hint next alpha chunk into GL2 (speculative, safe OOB)
      __builtin_prefetch(alpha + (size_t)(r64 + wave * 8) * N_NODES + k0 + KC, 0, 1);

#pragma unroll 4
    for (int k = 0; k < KC; k += 4) {
      const v2f af = { lds_a[16 * rg + mrow][k + kb],
                       lds_a[16 * rg + mrow][k + kb + 1] };
#pragma unroll
      for (int t = 0; t < 4; ++t) {
        const int n = cg * 64 + t * 16 + mrow;
        const v2f bf = { lds_w[k + kb][n], lds_w[k + kb + 1][n] };
        acc[t] = __builtin_amdgcn_wmma_f32_16x16x4_f32(false, af, false, bf,
                                                       (short)0, acc[t],
                                                       false, false);
      }
    }
    __syncthreads();
  }

  const int mb = (lane < 16) ? 0 : 8;
#pragma unroll
  for (int t = 0; t < 4; ++t) {
    const int n = cg * 64 + t * 16 + mrow;
#pragma unroll
    for (int p = 0; p < 8; ++p) {
      const int row = r64 + rg * 16 + mb + p;
      const float x = acc[t][p];
      z[(size_t)row * H_DIM + n] = 1.f / (1.f + __expf(-x));
    }
  }
}

// ---------------------------------------------------------------------------
extern "C" void kernel_launch(void* const* d_in, const int* in_sizes, int n_in,
                              void* d_out, int out_size, void* d_ws,
                              size_t ws_size, hipStream_t stream) {
  (void)in_sizes; (void)n_in; (void)out_size; (void)ws_size;
  const float* h   = (const float*)d_in[0];   // (8192, 512)
  const float* adj = (const float*)d_in[1];   // (8192, 8192)
  const float* Ww  = (const float*)d_in[2];   // (128, 512)
  const float* Wb  = (const float*)d_in[3];   // (128,)
  const float* a   = (const float*)d_in[4];   // (1, 128)

  float* z     = (float*)d_out;                         // (8192, 128)
  float* alpha = (float*)d_out + (size_t)N_NODES * H_DIM;  // (8192, 8192)

  float* Wh = (float*)d_ws;                             // 8192*128 f32
  float* s  = Wh + (size_t)N_NODES * H_DIM;             // 8192 f32

  wh_kernel<<<N_NODES / 16, 256, 0, stream>>>(h, Ww, Wb, a, Wh, s);
  alpha_kernel<<<N_NODES, 256, 0, stream>>>(adj, s, alpha);
  z_kernel<<<N_NODES / 64, 256, 0, stream>>>(alpha, Wh, z);
}